// HomographNodeEncoder_72327249264835
// MI455X (gfx1250) — compile-verified
//
#include <hip/hip_runtime.h>
#include <hip/hip_bf16.h>

// ---------------------------------------------------------------------------
// HomographNodeEncoder for MI455X (gfx1250, wave32).
//
// Strategy: fold the 4-type MoE projection into ONE uniform fp32 GEMM:
//   Wstack[64][256]: rows [16t .. 16t+13] = type-t weights scattered to their
//   source column positions (zero elsewhere -> discrete cols multiply by 0).
//   xe[row][64]:     zero except slots [16t, 16t+14) = x[row][0..13].
//   cont = xe @ Wstack  computed with V_WMMA_F32_16X16X4_F32 (exact fp32).
// Epilogue: branchless — per-row segment pointers precomputed once, selected
// per column with cndmask; bias + embedding gathers coalesce across lanes.
// Bounds check is hoisted to tile granularity: interior tiles (all but the
// last grid block) run a guard-free epilogue with unconditional NT stores.
// ---------------------------------------------------------------------------

typedef float v2f __attribute__((ext_vector_type(2)));
typedef float v8f __attribute__((ext_vector_type(8)));

#define EMB        256
#define WAVES      8          // 8 waves * 16 rows = 128 rows per block
#define ROWS_BLK   (WAVES * 16)
#define WT_STRIDE  68         // padded row stride (floats) -> conflict-free
#define XE_STRIDE  68

template <bool B> struct BoolC { static constexpr bool value = B; };

// inverse CONT maps: c_inv[t][c] = index i with CONT[t][i]==c, else -1
constexpr int c_inv[4][16] = {
    { 0, 1,-1,-1, 2,-1, 3, 4, 5, 6, 7, 8, 9,10,-1,-1},   // t=0, fan_in 11
    { 0, 1,-1,-1, 2, 3, 4, 5, 6, 7, 8, 9,10,11,-1,-1},   // t=1, fan_in 12
    {-1, 0, 1,-1, 2, 3, 4, 5, 6, 7, 8, 9,10,11,-1,-1},   // t=2, fan_in 12
    {-1,-1, 0, 1, 2, 3, 4, 5, 6, 7, 8, 9,10,11,-1,-1}    // t=3, fan_in 12
};

// Build WstackT[e][k] (k fastest, dense 64) in workspace:
//   WstackT[e][16t + c] = W_t[e][c_inv[t][c]]   (0 if c not in CONT[t])
__global__ void prep_kernel(const float* __restrict__ W0,
                            const float* __restrict__ W1,
                            const float* __restrict__ W2,
                            const float* __restrict__ W3,
                            float* __restrict__ WT)
{
    int e = threadIdx.x;               // 0..255 (one output channel each)
    #pragma unroll
    for (int t = 0; t < 4; ++t) {
        const float* W = (t == 0) ? W0 : (t == 1) ? W1 : (t == 2) ? W2 : W3;
        const int fan  = (t == 0) ? 11 : 12;
        #pragma unroll
        for (int c = 0; c < 16; ++c) {
            const int i = c_inv[t][c];                 // compile-time constant
            float v = 0.0f;
            if (i >= 0) v = W[e * fan + i];
            WT[e * 64 + 16 * t + c] = v;
        }
    }
}

__global__ __launch_bounds__(256)
void encoder_kernel(const float* __restrict__ x,
                    const int*   __restrict__ types,
                    const float* __restrict__ WTg,     // [256][64] from prep
                    const float* __restrict__ b0, const float* __restrict__ b1,
                    const float* __restrict__ b2, const float* __restrict__ b3,
                    const float* __restrict__ E0_2, const float* __restrict__ E0_3,
                    const float* __restrict__ E0_5,
                    const float* __restrict__ E1_2, const float* __restrict__ E1_3,
                    const float* __restrict__ E2_0,
                    const float* __restrict__ E3_0, const float* __restrict__ E3_1,
                    float* __restrict__ out, int N)
{
    extern __shared__ float smem[];
    float* wt = smem;                        // 256 * WT_STRIDE floats
    float* xe = smem + 256 * WT_STRIDE;      // WAVES * 16 * XE_STRIDE floats

    const int tid   = threadIdx.x;
    const int lane  = tid & 31;
    const int wave  = tid >> 5;
    const int khalf = lane >> 4;             // 0: K even pair, 1: K odd pair
    const int n     = lane & 15;             // column-in-tile for B/C/D

    // ---- stage WstackT into LDS (one column per thread, float4 copies) ----
    {
        const float4* src = (const float4*)(WTg + tid * 64);       // 16B aligned
        float4*       dst = (float4*)(wt + tid * WT_STRIDE);       // 272*tid: 16B
        #pragma unroll
        for (int q = 0; q < 16; ++q) dst[q] = src[q];
    }

    // ---- stage per-wave expanded inputs xe (zeros + 14-float window) ------
    const int rowBase = (blockIdx.x * WAVES + wave) * 16;
    float* myxe = xe + wave * 16 * XE_STRIDE;
    if (lane < 16) {
        const int r = rowBase + lane;
        float* dst = myxe + lane * XE_STRIDE;
        const float4 z = {0.f, 0.f, 0.f, 0.f};
        #pragma unroll
        for (int q = 0; q < 16; ++q) ((float4*)dst)[q] = z;        // k=0..63 := 0
        if (r < N) {
            const int t = types[r];
            const float2* xr = (const float2*)(x + (long long)r * 14); // 8B align
            float* w = dst + 16 * t;
            #pragma unroll
            for (int h = 0; h < 7; ++h) ((float2*)w)[h] = xr[h];   // 14 floats
        }
    }
    __syncthreads();

    // ---- A fragments: V_WMMA_F32_16X16X4_F32 layout ----------------------
    // lanes 0-15: M=lane, {K=4s, 4s+1}; lanes 16-31: M=lane-16, {K=4s+2, 4s+3}
    v2f afrag[16];
    {
        const float* arow = myxe + n * XE_STRIDE + 2 * khalf;
        #pragma unroll
        for (int s = 0; s < 16; ++s)
            afrag[s] = *(const v2f*)(arow + 4 * s);
    }

    // ---- per-row epilogue state, hoisted out of the j-loop ----------------
    // Segment pointers are pre-offset by (code*dim - seg_start) so the inner
    // loop is just: p = sel(col); val = acc + bias[col] + p[col].
    const float* pb[8];                       // bias base per row
    const float* P0[8]; const float* P1[8]; const float* P2[8];
    int th1[8], th2[8];
    #pragma unroll
    for (int v = 0; v < 8; ++v) {
        const int r  = rowBase + v + 8 * khalf;
        const int rc = (r < N) ? r : (N - 1);          // keeps pointers valid
        const int t  = types[rc];
        const float* xr = x + (long long)rc * 14;
        if (t == 0) {            // segs: [0,86) E0_2 | [86,171) E0_3 | E0_5
            pb[v] = b0;
            P0[v] = E0_2 + (int)xr[2] * 86;
            P1[v] = E0_3 + (int)xr[3] * 85 - 86;
            P2[v] = E0_5 + (int)xr[5] * 85 - 171;
            th1[v] = 86;  th2[v] = 171;
        } else if (t == 1) {     // segs: [0,128) E1_2 | [128,256) E1_3
            pb[v] = b1;
            P0[v] = E1_2 + (int)xr[2] * 128;
            P1[v] = E1_3 + (int)xr[3] * 128 - 128;
            P2[v] = P1[v];
            th1[v] = 128; th2[v] = 128;
        } else if (t == 2) {     // single seg E2_0
            pb[v] = b2;
            P0[v] = E2_0 + (int)xr[0] * 256;
            P1[v] = P0[v]; P2[v] = P0[v];
            th1[v] = 256; th2[v] = 256;
        } else {                 // segs: [0,128) E3_0 | [128,256) E3_1
            pb[v] = b3;
            P0[v] = E3_0 + (int)xr[0] * 128;
            P1[v] = E3_1 + (int)xr[1] * 128 - 128;
            P2[v] = P1[v];
            th1[v] = 128; th2[v] = 128;
        }
    }

    // ---- GEMM + fused branchless epilogue over 16 column tiles -----------
    // GUARD=false for interior tiles (3906 of 3907 blocks): no store guards.
    auto run_tiles = [&](auto guard_tag) {
        constexpr bool GUARD = decltype(guard_tag)::value;
        for (int j = 0; j < 16; ++j) {
            v8f acc0 = {}, acc1 = {};         // dual accumulators: break D->C chain
            const float* wcol = wt + (16 * j + n) * WT_STRIDE + 2 * khalf;
            #pragma unroll
            for (int s = 0; s < 16; s += 2) {
                v2f bf0 = *(const v2f*)(wcol + 4 * s);
                v2f bf1 = *(const v2f*)(wcol + 4 * s + 4);
                acc0 = __builtin_amdgcn_wmma_f32_16x16x4_f32(
                           false, afrag[s],     false, bf0, (short)0, acc0, false, false);
                acc1 = __builtin_amdgcn_wmma_f32_16x16x4_f32(
                           false, afrag[s + 1], false, bf1, (short)0, acc1, false, false);
            }

            const int col = 16 * j + n;
            #pragma unroll
            for (int v = 0; v < 8; ++v) {
                const int r = rowBase + v + 8 * khalf;    // C/D: M = v + 8*khalf
                const float* p = (col < th1[v]) ? P0[v]
                               : (col < th2[v]) ? P1[v] : P2[v];   // 2x cndmask_b64
                const float val = acc0[v] + acc1[v] + pb[v][col] + p[col];
                if (!GUARD || r < N)
                    __builtin_nontemporal_store(val, out + (long long)r * 256 + col);
            }
        }
    };

    if (rowBase + 16 <= N) run_tiles(BoolC<false>{});   // fast interior path
    else                   run_tiles(BoolC<true>{});    // tail tile only
}

extern "C" void kernel_launch(void* const* d_in, const int* in_sizes, int n_in,
                              void* d_out, int out_size, void* d_ws, size_t ws_size,
                              hipStream_t stream)
{
    (void)n_in; (void)out_size;
    const float* x     = (const float*)d_in[0];
    const int*   types = (const int*)  d_in[1];
    const float* W0 = (const float*)d_in[2];  const float* b0 = (const float*)d_in[3];
    const float* E0_2 = (const float*)d_in[4];
    const float* E0_3 = (const float*)d_in[5];
    const float* E0_5 = (const float*)d_in[6];
    const float* W1 = (const float*)d_in[7];  const float* b1 = (const float*)d_in[8];
    const float* E1_2 = (const float*)d_in[9];
    const float* E1_3 = (const float*)d_in[10];
    const float* W2 = (const float*)d_in[11]; const float* b2 = (const float*)d_in[12];
    const float* E2_0 = (const float*)d_in[13];
    const float* W3 = (const float*)d_in[14]; const float* b3 = (const float*)d_in[15];
    const float* E3_0 = (const float*)d_in[16];
    const float* E3_1 = (const float*)d_in[17];

    const int N = in_sizes[1];                 // node_types count = row count
    float* WTg = (float*)d_ws;                 // needs 64*256*4 = 64 KB scratch
    (void)ws_size;

    prep_kernel<<<1, 256, 0, stream>>>(W0, W1, W2, W3, WTg);

    const size_t smem = (size_t)(256 * WT_STRIDE + WAVES * 16 * XE_STRIDE) * sizeof(float);
    (void)hipFuncSetAttribute((const void*)encoder_kernel,
                              hipFuncAttributeMaxDynamicSharedMemorySize,
                              (int)smem);

    const int blocks = (N + ROWS_BLK - 1) / ROWS_BLK;
    encoder_kernel<<<blocks, 256, smem, stream>>>(
        x, types, WTg, b0, b1, b2, b3,
        E0_2, E0_3, E0_5, E1_2, E1_3, E2_0, E3_0, E3_1,
        (float*)d_out, N);
}